// GeometricConsistencyLoss_54631984005327
// MI455X (gfx1250) — compile-verified
//
#include <hip/hip_runtime.h>
#include <hip/hip_bf16.h>

#define LBL 32
#define NB 16
#define NPTS 262144
#define BLOCKS_PER_BATCH 32
#define CHUNK (NPTS / BLOCKS_PER_BATCH)   // 8192
#define WAVES_PER_BLOCK 8
#define PPW (CHUNK / WAVES_PER_BLOCK)     // 1024 points per wave
#define ITERS (PPW / 32)                  // 32 iterations of 32 points
#define NSEG (NB * LBL)                   // 512

typedef float v2f __attribute__((ext_vector_type(2)));
typedef float v8f __attribute__((ext_vector_type(8)));

// ---------------------------------------------------------------------------
// Kernel 0: zero workspace accumulators
// ---------------------------------------------------------------------------
__global__ void gc_zero(float* p, int n) {
    int i = blockIdx.x * blockDim.x + threadIdx.x;
    if (i < n) p[i] = 0.0f;
}

// ---------------------------------------------------------------------------
// Kernel 1: pass 1 — segmented {count, sum, outer, nsum} via one-hot WMMA GEMM
// gstats[seg][16] = { cnt, sx, sy, sz, xx, xy, xz, yy, yz, zz, nx, ny, nz, 0,0,0 }
//
// The feat/labs LDS tiles are PER-WAVE private: cross-lane visibility within a
// wave only needs the hardware's in-order LDS pipeline (DScnt, in-order) plus a
// compiler scheduling fence (wave_barrier) — no workgroup barrier per iteration.
// ---------------------------------------------------------------------------
__global__ __launch_bounds__(256)
void gc_pass1(const float* __restrict__ pts, const int* __restrict__ lab,
              const float* __restrict__ nrm, float* __restrict__ gstats) {
    __shared__ float feat[WAVES_PER_BLOCK][32][16];
    __shared__ int   labs[WAVES_PER_BLOCK][32];
    __shared__ float bstats[LBL][16];

    const int tid  = threadIdx.x;
    const int wave = tid >> 5;
    const int lane = tid & 31;
    const int b    = blockIdx.x / BLOCKS_PER_BATCH;
    const int blk  = blockIdx.x % BLOCKS_PER_BATCH;
    const int base = b * NPTS + blk * CHUNK + wave * PPW;

    for (int i = tid; i < LBL * 16; i += 256) ((float*)bstats)[i] = 0.0f;
    __syncthreads();

    v8f acc0 = {};           // labels 0..15  x features 0..15
    v8f acc1 = {};           // labels 16..31 x features 0..15
    const int mrow  = lane & 15;              // matrix row / column index
    const int khalf = (lane >> 4) ? 2 : 0;    // K pair owned by this lane half

    for (int it = 0; it < ITERS; ++it) {
        const int n = base + it * 32 + lane;
        const int l = lab[n];
        const float x  = pts[n * 3 + 0], y  = pts[n * 3 + 1], z  = pts[n * 3 + 2];
        const float nx = nrm[n * 3 + 0], ny = nrm[n * 3 + 1], nz = nrm[n * 3 + 2];

        float4* fv = (float4*)&feat[wave][lane][0];
        float4 f0 = {1.0f, x, y, z};
        float4 f1 = {x * x, x * y, x * z, y * y};
        float4 f2 = {y * z, z * z, nx, ny};
        float4 f3 = {nz, 0.0f, 0.0f, 0.0f};
        fv[0] = f0; fv[1] = f1; fv[2] = f2; fv[3] = f3;
        labs[wave][lane] = l;
        // Per-wave LDS region: hardware keeps this wave's DS ops in order, so
        // our subsequent loads see these stores. Only block compiler reordering.
        __builtin_amdgcn_wave_barrier();

        #pragma unroll
        for (int g = 0; g < 8; ++g) {
            // B (4x16): this lane holds B[khalf][mrow], B[khalf+1][mrow]
            v2f bv;
            bv.x = feat[wave][g * 4 + khalf + 0][mrow];
            bv.y = feat[wave][g * 4 + khalf + 1][mrow];
            const int l0 = labs[wave][g * 4 + khalf + 0];
            const int l1 = labs[wave][g * 4 + khalf + 1];
            // A (16x4) one-hot of labels 0..15
            v2f a0;
            a0.x = (l0 == mrow) ? 1.0f : 0.0f;
            a0.y = (l1 == mrow) ? 1.0f : 0.0f;
            acc0 = __builtin_amdgcn_wmma_f32_16x16x4_f32(
                false, a0, false, bv, (short)0, acc0, false, false);
            // A one-hot of labels 16..31
            v2f a1;
            a1.x = (l0 == mrow + 16) ? 1.0f : 0.0f;
            a1.y = (l1 == mrow + 16) ? 1.0f : 0.0f;
            acc1 = __builtin_amdgcn_wmma_f32_16x16x4_f32(
                false, a1, false, bv, (short)0, acc1, false, false);
        }
        // Keep next iteration's stores from being scheduled above these loads.
        __builtin_amdgcn_wave_barrier();
    }

    // D layout: VGPR r, lanes 0-15 -> row r, col=lane; lanes 16-31 -> row r+8
    const int hi = (lane >> 4) ? 8 : 0;
    #pragma unroll
    for (int r = 0; r < 8; ++r) {
        atomicAdd(&bstats[r + hi][mrow],      acc0[r]);
        atomicAdd(&bstats[r + hi + 16][mrow], acc1[r]);
    }
    __syncthreads();

    for (int i = tid; i < LBL * 16; i += 256)
        atomicAdd(&gstats[b * LBL * 16 + i], ((float*)bstats)[i]);
}

// ---------------------------------------------------------------------------
// Kernel 2: per-segment covariance + 3x3 Jacobi eigensolve
// gseg[seg][8] = { n.x, n.y, n.z, d, mn.x, mn.y, mn.z, cnt }
// ---------------------------------------------------------------------------
__global__ void gc_solve(const float* __restrict__ gstats, float* __restrict__ gseg) {
    const int s = blockIdx.x * blockDim.x + threadIdx.x;
    if (s >= NSEG) return;
    const float* st = &gstats[s * 16];
    const float count = st[0];
    const float cnt   = fmaxf(count, 1.0f);
    const float cx = st[1] / cnt, cy = st[2] / cnt, cz = st[3] / cnt;
    float A[3][3];
    A[0][0] = st[4] - count * cx * cx;
    A[0][1] = st[5] - count * cx * cy;
    A[0][2] = st[6] - count * cx * cz;
    A[1][1] = st[7] - count * cy * cy;
    A[1][2] = st[8] - count * cy * cz;
    A[2][2] = st[9] - count * cz * cz;
    A[1][0] = A[0][1]; A[2][0] = A[0][2]; A[2][1] = A[1][2];

    const bool valid = (count >= 50.0f) && ((s & (LBL - 1)) != 0);
    if (!valid) {
        A[0][0] = A[1][1] = A[2][2] = 1.0f;
        A[0][1] = A[0][2] = A[1][2] = A[1][0] = A[2][0] = A[2][1] = 0.0f;
    }

    float V[3][3] = {{1, 0, 0}, {0, 1, 0}, {0, 0, 1}};
    for (int sweep = 0; sweep < 12; ++sweep) {
        for (int pq = 0; pq < 3; ++pq) {
            const int p = (pq == 2) ? 1 : 0;
            const int q = (pq == 0) ? 1 : 2;
            const float apq = A[p][q];
            if (fabsf(apq) > 1e-20f) {
                const float theta = (A[q][q] - A[p][p]) / (2.0f * apq);
                const float t = copysignf(1.0f, theta) /
                                (fabsf(theta) + sqrtf(theta * theta + 1.0f));
                const float c  = rsqrtf(t * t + 1.0f);
                const float sn = t * c;
                for (int k = 0; k < 3; ++k) {
                    const float akp = A[k][p], akq = A[k][q];
                    A[k][p] = c * akp - sn * akq;
                    A[k][q] = sn * akp + c * akq;
                }
                for (int k = 0; k < 3; ++k) {
                    const float apk = A[p][k], aqk = A[q][k];
                    A[p][k] = c * apk - sn * aqk;
                    A[q][k] = sn * apk + c * aqk;
                }
                for (int k = 0; k < 3; ++k) {
                    const float vkp = V[k][p], vkq = V[k][q];
                    V[k][p] = c * vkp - sn * vkq;
                    V[k][q] = sn * vkp + c * vkq;
                }
            }
        }
    }
    int m = 0;
    if (A[1][1] < A[0][0]) m = 1;
    if (A[2][2] < A[m][m]) m = 2;
    const float nx = V[0][m], ny = V[1][m], nz = V[2][m];
    const float d  = -(nx * cx + ny * cy + nz * cz);

    float* o = &gseg[s * 8];
    o[0] = nx; o[1] = ny; o[2] = nz; o[3] = d;
    o[4] = st[10] / cnt; o[5] = st[11] / cnt; o[6] = st[12] / cnt;
    o[7] = cnt;
}

// ---------------------------------------------------------------------------
// Kernel 3: pass 2 — dist_sum and dev_sum per segment
// gpair[seg][2] = { dist_sum, dev_sum }
// ---------------------------------------------------------------------------
__global__ __launch_bounds__(256)
void gc_pass2(const float* __restrict__ pts, const int* __restrict__ lab,
              const float* __restrict__ nrm, const float* __restrict__ gseg,
              float* __restrict__ gpair) {
    __shared__ float seg[LBL][8];
    __shared__ float acc[LBL][2];
    const int tid = threadIdx.x;
    const int b   = blockIdx.x / BLOCKS_PER_BATCH;
    const int blk = blockIdx.x % BLOCKS_PER_BATCH;

    for (int i = tid; i < LBL * 8; i += 256)
        ((float*)seg)[i] = gseg[b * LBL * 8 + i];
    for (int i = tid; i < LBL * 2; i += 256)
        ((float*)acc)[i] = 0.0f;
    __syncthreads();

    const int base = b * NPTS + blk * CHUNK;
    for (int n = base + tid; n < base + CHUNK; n += 256) {
        const int l = lab[n];
        const float x  = pts[n * 3 + 0], y  = pts[n * 3 + 1], z  = pts[n * 3 + 2];
        const float nx = nrm[n * 3 + 0], ny = nrm[n * 3 + 1], nz = nrm[n * 3 + 2];
        const float dist = fabsf(x * seg[l][0] + y * seg[l][1] + z * seg[l][2] + seg[l][3]);
        const float dev  = fabsf(nx - seg[l][4]) + fabsf(ny - seg[l][5]) + fabsf(nz - seg[l][6]);
        atomicAdd(&acc[l][0], dist);
        atomicAdd(&acc[l][1], dev);
    }
    __syncthreads();
    for (int i = tid; i < LBL * 2; i += 256)
        atomicAdd(&gpair[b * LBL * 2 + i], ((float*)acc)[i]);
}

// ---------------------------------------------------------------------------
// Kernel 4: final scalar reduction with masking
// ---------------------------------------------------------------------------
__global__ __launch_bounds__(512)
void gc_final(const float* __restrict__ gstats, const float* __restrict__ gpair,
              float* __restrict__ out) {
    __shared__ float sh0[512], sh1[512], sh2[512];
    const int s = threadIdx.x;            // one thread per segment (512)
    const float count = gstats[s * 16];
    const float cnt   = fmaxf(count, 1.0f);
    const bool  nz      = (s & (LBL - 1)) != 0;
    const bool  valid   = (count >= 50.0f) && nz;
    const bool  valid_s = (count >= 2.0f) && nz;
    sh0[s] = valid   ? gpair[s * 2 + 0] / cnt          : 0.0f;
    sh1[s] = valid   ? 1.0f                            : 0.0f;
    sh2[s] = valid_s ? gpair[s * 2 + 1] / (cnt * 3.0f) : 0.0f;
    __syncthreads();
    for (int off = 256; off > 0; off >>= 1) {
        if (s < off) {
            sh0[s] += sh0[s + off];
            sh1[s] += sh1[s + off];
            sh2[s] += sh2[s + off];
        }
        __syncthreads();
    }
    if (s == 0) {
        const float np    = sh1[0];
        const float total = sh0[0] / fmaxf(np, 1.0f) + 0.1f * sh2[0];
        out[0] = (np > 0.0f) ? total : 0.0f;
    }
}

// ---------------------------------------------------------------------------
extern "C" void kernel_launch(void* const* d_in, const int* in_sizes, int n_in,
                              void* d_out, int out_size, void* d_ws, size_t ws_size,
                              hipStream_t stream) {
    const float* pts = (const float*)d_in[0];
    const int*   lab = (const int*)d_in[1];
    const float* nrm = (const float*)d_in[2];
    float* out = (float*)d_out;

    float* gstats = (float*)d_ws;             // NSEG * 16 floats
    float* gseg   = gstats + NSEG * 16;       // NSEG * 8  floats
    float* gpair  = gseg + NSEG * 8;          // NSEG * 2  floats
    const int nzero = NSEG * (16 + 8 + 2);

    gc_zero<<<(nzero + 255) / 256, 256, 0, stream>>>((float*)d_ws, nzero);
    gc_pass1<<<NB * BLOCKS_PER_BATCH, 256, 0, stream>>>(pts, lab, nrm, gstats);
    gc_solve<<<(NSEG + 255) / 256, 256, 0, stream>>>(gstats, gseg);
    gc_pass2<<<NB * BLOCKS_PER_BATCH, 256, 0, stream>>>(pts, lab, nrm, gseg, gpair);
    gc_final<<<1, 512, 0, stream>>>(gstats, gpair, out);
}